// AdaptiveTokenMerger_77773267796181
// MI455X (gfx1250) — compile-verified
//
#include <hip/hip_runtime.h>
#include <hip/hip_bf16.h>

// ---------------- problem constants ----------------
#define BATCH 128
#define NTOK  196
#define DIM   768
#define NTILE 13                         // ceil(196/16) row tiles
#define NCGRP 4                          // column groups of 4 tiles (64 cols)
#define JOBS_PER_BATCH (NTILE * NCGRP)   // 52 wave-jobs
#define BLOCKS_PER_BATCH 7               // 7 blocks * 8 waves = 56 >= 52
#define SIM_THRESH 0.9f
#define MOTION_SCALE 0.5f
#define SALIENCY_SCALE 0.5f
#define EPSV 1e-6f

// output layout (flat float elements in d_out, return order)
#define MERGED_OFF 0
#define MERGED_SZ  ((size_t)BATCH * NTOK * DIM)
#define SIM_OFF    (MERGED_OFF + MERGED_SZ)
#define SIM_SZ     ((size_t)BATCH * NTOK * NTOK)
#define IMP_OFF    (SIM_OFF + SIM_SZ)
#define IMP_SZ     ((size_t)BATCH * NTOK)
#define GID_OFF    (IMP_OFF + IMP_SZ)

typedef __attribute__((ext_vector_type(2))) float v2f;
typedef __attribute__((ext_vector_type(4))) float v4f;
typedef __attribute__((ext_vector_type(8))) float v8f;

#define WMMA_F32(A, B, C) \
    __builtin_amdgcn_wmma_f32_16x16x4_f32(false, (A), false, (B), (short)0, (C), false, false)

// ---------------- kernel 0: zero merged region + global maxima ----------------
__global__ __launch_bounds__(256) void init_kernel(float* merged, unsigned* gmax) {
    size_t i = (size_t)blockIdx.x * blockDim.x + threadIdx.x;
    size_t stride = (size_t)gridDim.x * blockDim.x;
    for (size_t k = i; k < MERGED_SZ; k += stride) merged[k] = 0.0f;
    if (i < 2) gmax[i] = 0u;
}

// ---------------- kernel 1: per-token inverse L2 norm ----------------
__global__ __launch_bounds__(256) void norm_kernel(const float* __restrict__ tokens,
                                                   float* __restrict__ invnorm) {
    const int t = blockIdx.x;                 // token index over B*N
    const float* p = tokens + (size_t)t * DIM;
    float s = 0.0f;
    for (int d = threadIdx.x; d < DIM; d += 256) { float v = p[d]; s += v * v; }
    for (int o = 16; o; o >>= 1) s += __shfl_down(s, o);
    __shared__ float ls[8];
    if ((threadIdx.x & 31) == 0) ls[threadIdx.x >> 5] = s;
    __syncthreads();
    if (threadIdx.x == 0) {
        float tot = 0.0f;
        #pragma unroll
        for (int i = 0; i < 8; i++) tot += ls[i];
        float nrm = sqrtf(tot);
        invnorm[t] = 1.0f / fmaxf(nrm, 1e-12f);
    }
}

// ---------------- kernel 2: global max of motion & saliency ----------------
__global__ __launch_bounds__(256) void max_kernel(const float* __restrict__ motion,
                                                  const float* __restrict__ saliency,
                                                  unsigned* __restrict__ gmax, int n) {
    float mm = 0.0f, ss = 0.0f;   // inputs are uniform[0,1): non-negative
    for (int i = blockIdx.x * blockDim.x + threadIdx.x; i < n;
         i += gridDim.x * blockDim.x) {
        mm = fmaxf(mm, motion[i]);
        ss = fmaxf(ss, saliency[i]);
    }
    for (int o = 16; o; o >>= 1) {
        mm = fmaxf(mm, __shfl_down(mm, o));
        ss = fmaxf(ss, __shfl_down(ss, o));
    }
    if ((threadIdx.x & 31) == 0) {
        atomicMax(&gmax[0], __float_as_uint(mm));   // uint order == float order (>=0)
        atomicMax(&gmax[1], __float_as_uint(ss));
    }
}

// ---------------- kernel 3: importance per batch row ----------------
__global__ __launch_bounds__(256) void imp_kernel(const float* __restrict__ motion,
                                                  const float* __restrict__ saliency,
                                                  const float* __restrict__ ratio,
                                                  const unsigned* __restrict__ gmax,
                                                  float* __restrict__ imp_out) {
    const int b = blockIdx.x;
    const int i = threadIdx.x;
    const bool valid = i < NTOK;
    const float M = __uint_as_float(gmax[0]) + EPSV;
    const float S = __uint_as_float(gmax[1]) + EPSV;
    float x = 0.0f;
    if (valid) {
        float m = motion[b * NTOK + i] / M;
        float s = saliency[b * NTOK + i] / S;
        float r = ratio[b * NTOK + i];
        const float rc = 1.0f - MOTION_SCALE - SALIENCY_SCALE;  // == 0.0
        x = MOTION_SCALE * m + SALIENCY_SCALE * s + rc * r;
    }
    float lo = valid ? x : 3.4e38f;
    float hi = valid ? x : -3.4e38f;
    for (int o = 16; o; o >>= 1) {
        lo = fminf(lo, __shfl_down(lo, o));
        hi = fmaxf(hi, __shfl_down(hi, o));
    }
    __shared__ float slo[8], shi[8];
    __shared__ float LO, HI;
    if ((threadIdx.x & 31) == 0) { slo[threadIdx.x >> 5] = lo; shi[threadIdx.x >> 5] = hi; }
    __syncthreads();
    if (threadIdx.x == 0) {
        float l = slo[0], h = shi[0];
        #pragma unroll
        for (int k = 1; k < 8; k++) { l = fminf(l, slo[k]); h = fmaxf(h, shi[k]); }
        LO = l; HI = h;
    }
    __syncthreads();
    if (valid) imp_out[b * NTOK + i] = (x - LO) / (HI - LO + EPSV);
}

// ---------------- kernel 4: WMMA f32 similarity GEMM (sim = Tn * Tn^T) ----------------
// One wave computes a 16x64 output strip (4 column tiles). Normalization is applied
// once to the final accumulator. Fragments are loaded 16B at a time (b128): one load
// feeds two WMMA K-steps (.xy / .zw), giving 5 loads per 8 WMMAs. The K->slot mapping
// (lanes 16-31 offset by +4 within each 8-K chunk) is identical for A and B, so the
// dot product is exact up to summation order.
__global__ __launch_bounds__(256) void sim_wmma_kernel(const float* __restrict__ tokens,
                                                       const float* __restrict__ invnorm,
                                                       float* __restrict__ sim) {
    const int wave = threadIdx.x >> 5;
    const int lane = threadIdx.x & 31;
    const int b    = blockIdx.x / BLOCKS_PER_BATCH;
    const int job  = (blockIdx.x % BLOCKS_PER_BATCH) * 8 + wave;
    if (job >= JOBS_PER_BATCH) return;            // wave-uniform: EXEC stays all-1s

    const int trow = job / NCGRP;                 // 0..12
    const int cg   = job % NCGRP;                 // 0..3
    const int row0 = trow * 16;
    const int colbase = cg * 64;

    const float* tb = tokens + (size_t)b * NTOK * DIM;
    const float* nb = invnorm + b * NTOK;

    const int mi  = lane & 15;
    const int kb4 = (lane >> 4) * 4;              // half-wave K offset within 8-K chunk

    const int arow = row0 + mi;
    const float* aptr = tb + (size_t)(arow < NTOK ? arow : 0) * DIM + kb4;

    const int c0 = colbase + mi;
    const int c1 = colbase + 16 + mi;
    const int c2 = colbase + 32 + mi;
    const int c3 = colbase + 48 + mi;
    const float* bptr0 = tb + (size_t)(c0 < NTOK ? c0 : 0) * DIM + kb4;
    const float* bptr1 = tb + (size_t)(c1 < NTOK ? c1 : 0) * DIM + kb4;
    const float* bptr2 = tb + (size_t)(c2 < NTOK ? c2 : 0) * DIM + kb4;
    const float* bptr3 = tb + (size_t)(c3 < NTOK ? c3 : 0) * DIM + kb4;

    v8f acc0 = {}, acc1 = {}, acc2 = {}, acc3 = {};

    #pragma unroll 2
    for (int k = 0; k < DIM; k += 8) {
        const v4f A  = *(const v4f*)(aptr  + k);
        const v4f B0 = *(const v4f*)(bptr0 + k);
        const v4f B1 = *(const v4f*)(bptr1 + k);
        const v4f B2 = *(const v4f*)(bptr2 + k);
        const v4f B3 = *(const v4f*)(bptr3 + k);
        acc0 = WMMA_F32(A.xy, B0.xy, acc0);
        acc1 = WMMA_F32(A.xy, B1.xy, acc1);
        acc2 = WMMA_F32(A.xy, B2.xy, acc2);
        acc3 = WMMA_F32(A.xy, B3.xy, acc3);
        acc0 = WMMA_F32(A.zw, B0.zw, acc0);
        acc1 = WMMA_F32(A.zw, B1.zw, acc1);
        acc2 = WMMA_F32(A.zw, B2.zw, acc2);
        acc3 = WMMA_F32(A.zw, B3.zw, acc3);
    }

    // C/D layout: VGPR v, lanes 0-15 -> M=v,   N=lane;
    //                     lanes 16-31 -> M=v+8, N=lane-16
    const int nloc  = lane & 15;
    const int mbase = (lane >> 4) * 8;

    float arsc[8];
    #pragma unroll
    for (int v = 0; v < 8; v++) {
        const int gm = row0 + mbase + v;
        arsc[v] = (gm < NTOK) ? nb[gm] : 0.0f;
    }

    v8f accs[4] = {acc0, acc1, acc2, acc3};
    float* simb = sim + (size_t)b * NTOK * NTOK;
    #pragma unroll
    for (int t = 0; t < 4; t++) {
        const int gn = colbase + t * 16 + nloc;
        if (gn >= NTOK) continue;
        const float bs = nb[gn];
        #pragma unroll
        for (int v = 0; v < 8; v++) {
            const int gm = row0 + mbase + v;
            if (gm < NTOK)
                simb[(size_t)gm * NTOK + gn] =
                    (gm == gn) ? 0.0f : accs[t][v] * arsc[v] * bs;
        }
    }
}

// ---------------- kernel 5: sequential-root BFS grouping (per batch) ----------------
__global__ __launch_bounds__(256) void group_kernel(const float* __restrict__ sim,
                                                    const float* __restrict__ imp,
                                                    int* __restrict__ gids_out,
                                                    float* __restrict__ w_out,
                                                    int* __restrict__ memlist,
                                                    int* __restrict__ gstart,
                                                    int* __restrict__ gcount) {
    const int b = blockIdx.x;
    __shared__ unsigned adj[NTOK][7];
    __shared__ unsigned reach[7], expanded[7], assigned[7], allowed[7];
    __shared__ int changed, ng;
    __shared__ int gids[NTOK], cnt[NTOK], startA[NTOK], cursor[NTOK];
    __shared__ float denom[NTOK];

    const float* simb = sim + (size_t)b * NTOK * NTOK;
    const float* impb = imp + b * NTOK;

    // adjacency bitset: edge i->j iff sim[i][j] > 0.9 && (1 - imp[i]) > 0.5
    for (int idx = threadIdx.x; idx < NTOK * 7; idx += blockDim.x) {
        const int i = idx / 7, w = idx % 7;
        unsigned bits = 0u;
        if ((1.0f - impb[i]) > 0.5f) {
            const float* srow = simb + (size_t)i * NTOK;
            for (int jb = 0; jb < 32; jb++) {
                const int j = w * 32 + jb;
                if (j < NTOK && srow[j] > SIM_THRESH) bits |= (1u << jb);
            }
        }
        adj[i][w] = bits;
    }
    if (threadIdx.x < 7) assigned[threadIdx.x] = 0u;
    if (threadIdx.x == 0) ng = 0;
    __syncthreads();

    for (int r = 0; r < NTOK; r++) {
        __syncthreads();
        if ((assigned[r >> 5] >> (r & 31)) & 1u) continue;   // uniform branch
        if (threadIdx.x < 7) {
            allowed[threadIdx.x]  = ~assigned[threadIdx.x];
            reach[threadIdx.x]    = 0u;
            expanded[threadIdx.x] = 0u;
        }
        __syncthreads();
        if (threadIdx.x == 0) {
            allowed[r >> 5] |= (1u << (r & 31));
            reach[r >> 5]   |= (1u << (r & 31));
        }
        __syncthreads();
        while (true) {
            if (threadIdx.x == 0) changed = 0;
            __syncthreads();
            for (int i = threadIdx.x; i < NTOK; i += blockDim.x) {
                const unsigned bit = 1u << (i & 31);
                if ((reach[i >> 5] & bit) && !(expanded[i >> 5] & bit)) {
                    atomicOr(&expanded[i >> 5], bit);
                    #pragma unroll
                    for (int w = 0; w < 7; w++) {
                        const unsigned add = adj[i][w] & allowed[w];
                        if (add) {
                            const unsigned old = atomicOr(&reach[w], add);
                            if (add & ~old) changed = 1;
                        }
                    }
                }
            }
            __syncthreads();
            if (!changed) break;
        }
        for (int i = threadIdx.x; i < NTOK; i += blockDim.x)
            if ((reach[i >> 5] >> (i & 31)) & 1u) gids[i] = ng;
        if (threadIdx.x < 7) atomicOr(&assigned[threadIdx.x], reach[threadIdx.x]);
        __syncthreads();
        if (threadIdx.x == 0) ng++;
    }
    __syncthreads();

    // segment denominators, member lists, weights
    for (int g = threadIdx.x; g < NTOK; g += blockDim.x) { denom[g] = 0.0f; cnt[g] = 0; }
    __syncthreads();
    for (int i = threadIdx.x; i < NTOK; i += blockDim.x) {
        atomicAdd(&denom[gids[i]], impb[i]);
        atomicAdd(&cnt[gids[i]], 1);
    }
    __syncthreads();
    if (threadIdx.x == 0) {
        int s = 0;
        for (int g = 0; g < NTOK; g++) { startA[g] = s; cursor[g] = s; s += cnt[g]; }
    }
    __syncthreads();
    for (int i = threadIdx.x; i < NTOK; i += blockDim.x) {
        const int pos = atomicAdd(&cursor[gids[i]], 1);
        memlist[b * NTOK + pos] = i;
        gids_out[b * NTOK + i] = gids[i];
        w_out[b * NTOK + i] = impb[i] / (denom[gids[i]] + EPSV);
    }
    for (int g = threadIdx.x; g < NTOK; g += blockDim.x) {
        gstart[b * NTOK + g] = startA[g];
        gcount[b * NTOK + g] = cnt[g];
    }
}

// ---------------- kernel 6: weighted merge per (batch, group) ----------------
__global__ __launch_bounds__(256) void merge_kernel(const float* __restrict__ tokens,
                                                    const float* __restrict__ w,
                                                    const int* __restrict__ memlist,
                                                    const int* __restrict__ gstart,
                                                    const int* __restrict__ gcount,
                                                    float* __restrict__ merged) {
    const int b = blockIdx.x / NTOK;
    const int g = blockIdx.x % NTOK;
    const int c = gcount[b * NTOK + g];
    if (c == 0) return;                        // empty groups stay zero (pre-zeroed)
    const int s0 = gstart[b * NTOK + g];
    const float* tb = tokens + (size_t)b * NTOK * DIM;
    float* out = merged + ((size_t)b * NTOK + g) * DIM;
    float acc0 = 0.0f, acc1 = 0.0f, acc2 = 0.0f;   // 768 / 256 = 3 per thread
    for (int m = 0; m < c; m++) {
        const int i = memlist[b * NTOK + s0 + m];
        const float wi = w[b * NTOK + i];
        const float* tp = tb + (size_t)i * DIM;
        acc0 += tp[threadIdx.x       ] * wi;
        acc1 += tp[threadIdx.x +  256] * wi;
        acc2 += tp[threadIdx.x +  512] * wi;
    }
    out[threadIdx.x      ] = acc0;
    out[threadIdx.x + 256] = acc1;
    out[threadIdx.x + 512] = acc2;
}

// ---------------- launcher ----------------
extern "C" void kernel_launch(void* const* d_in, const int* in_sizes, int n_in,
                              void* d_out, int out_size, void* d_ws, size_t ws_size,
                              hipStream_t stream) {
    const float* tokens   = (const float*)d_in[0];
    const float* motion   = (const float*)d_in[1];
    const float* saliency = (const float*)d_in[2];
    const float* ratio    = (const float*)d_in[3];

    float* out    = (float*)d_out;
    float* merged = out + MERGED_OFF;
    float* sim    = out + SIM_OFF;
    float* impo   = out + IMP_OFF;
    int*   gidso  = (int*)(out + GID_OFF);

    // workspace layout
    unsigned* gmax  = (unsigned*)d_ws;
    float* invnorm  = (float*)d_ws + 16;
    float* wgt      = invnorm + BATCH * NTOK;
    int*   memlist  = (int*)(wgt + BATCH * NTOK);
    int*   gstartp  = memlist + BATCH * NTOK;
    int*   gcountp  = gstartp + BATCH * NTOK;

    init_kernel<<<2048, 256, 0, stream>>>(merged, gmax);
    norm_kernel<<<BATCH * NTOK, 256, 0, stream>>>(tokens, invnorm);
    max_kernel<<<128, 256, 0, stream>>>(motion, saliency, gmax, BATCH * NTOK);
    imp_kernel<<<BATCH, 256, 0, stream>>>(motion, saliency, ratio, gmax, impo);
    sim_wmma_kernel<<<BATCH * BLOCKS_PER_BATCH, 256, 0, stream>>>(tokens, invnorm, sim);
    group_kernel<<<BATCH, 256, 0, stream>>>(sim, impo, gidso, wgt, memlist, gstartp, gcountp);
    merge_kernel<<<BATCH * NTOK, 256, 0, stream>>>(tokens, wgt, memlist, gstartp, gcountp, merged);
}